// CausalSelfAttention_46935402611424
// MI455X (gfx1250) — compile-verified
//
#include <hip/hip_runtime.h>
#include <hip/hip_bf16.h>
#include <stdint.h>

// ---------------------------------------------------------------------------
// CausalSelfAttention (GQA, int8 K-cache, RoPE) for MI455X / gfx1250, wave32.
// All GEMMs run on the WMMA pipe with V_WMMA_F32_16X16X4_F32 (fp32 exact).
// GEMM tile staging uses GLOBAL_LOAD_ASYNC_TO_LDS_B128 (ASYNCcnt-tracked DMA).
// ---------------------------------------------------------------------------

#define B_   2
#define T_   1024
#define C_   4096
#define NH   32
#define NKV  8
#define HD   128
#define P_   1024
#define S_   2048      // P_ + T_
#define KVC  2048      // 2 * NKV * HD

typedef __attribute__((ext_vector_type(2))) float v2f;
typedef __attribute__((ext_vector_type(8))) float v8f;

#if defined(__has_builtin)
#if __has_builtin(__builtin_amdgcn_global_load_async_to_lds_b128)
#define HAVE_ASYNC_LDS 1
#endif
#endif
#ifndef HAVE_ASYNC_LDS
#define HAVE_ASYNC_LDS 0
#endif

// Pointer types matching the builtin's signature (from hipcc diagnostic):
// param0: v4i in addrspace(1) (printed "int __device__ *"), param1: LDS v4i.
typedef int v4i_raw __attribute__((vector_size(4 * sizeof(int))));
typedef __attribute__((address_space(1))) v4i_raw* g_v4i_ptr;
typedef __attribute__((address_space(3))) v4i_raw* l_v4i_ptr;

// 16-byte global -> LDS copy: async DMA path on gfx1250 if declared,
// otherwise VGPR round-trip.
__device__ __forceinline__ void cp16_g2l(const float* g, float* l) {
#if HAVE_ASYNC_LDS
  __builtin_amdgcn_global_load_async_to_lds_b128((g_v4i_ptr)g, (l_v4i_ptr)l, 0,
                                                 0);
#else
  *reinterpret_cast<float4*>(l) = *reinterpret_cast<const float4*>(g);
#endif
}

__device__ __forceinline__ void wait_async0() {
#if HAVE_ASYNC_LDS
#if __has_builtin(__builtin_amdgcn_s_wait_asynccnt)
  __builtin_amdgcn_s_wait_asynccnt(0);
#else
  asm volatile("s_wait_asynccnt 0x0" ::: "memory");
#endif
#endif
}

__device__ __forceinline__ v8f wmma_f32_k4(v2f a, v2f b, v8f c) {
  // 8 args: (neg_a, A, neg_b, B, c_mod, C, reuse_a, reuse_b)
  return __builtin_amdgcn_wmma_f32_16x16x4_f32(false, a, false, b, (short)0, c,
                                               false, false);
}

// ---------------------------------------------------------------------------
// Tiled fp32 GEMM: C[M,N] = A[M,K] * B[K,N], row-major.
// Block = 128 threads (4 waves). Block tile 128x64, K-step 32.
// Wave w computes rows [32w,32w+32) x 64 cols as 2x4 WMMA C-tiles; each B
// fragment is reused for two A rows (8 WMMA per K-step-of-4).
// ---------------------------------------------------------------------------
__global__ __launch_bounds__(128) void gemm_f32_wmma(
    const float* __restrict__ A, const float* __restrict__ Bm,
    float* __restrict__ Cm, int M, int N, int K) {
  __shared__ float As[128 * 36];  // 128x32, padded stride 36 (18.0 KB)
  __shared__ float Bs[32 * 68];   // 32x64, padded stride 68 (8.5 KB)
  const int bm = blockIdx.y * 128, bn = blockIdx.x * 64;
  const int tid = threadIdx.x;
  const int wave = tid >> 5, lane = tid & 31;
  const int rowl = lane & 15;
  const int kh = (lane >> 4) << 1;  // K sub-offset per half-wave (frag layout)

  v8f acc0[4] = {};  // rows 32w + rowl group
  v8f acc1[4] = {};  // rows 32w + 16 + rowl group

  for (int k0 = 0; k0 < K; k0 += 32) {
    // ---- stage A (128x32) and B (32x64) via async copies to LDS ----
#pragma unroll
    for (int i = 0; i < 8; i++) {
      const int idx = tid + i * 128;  // A float4 index 0..1023
      const int ar = idx >> 3, ac = (idx & 7) << 2;
      cp16_g2l(A + (size_t)(bm + ar) * K + k0 + ac, &As[ar * 36 + ac]);
    }
#pragma unroll
    for (int i = 0; i < 4; i++) {
      const int idx = tid + i * 128;  // B float4 index 0..511
      const int br = idx >> 4, bc = (idx & 15) << 2;
      cp16_g2l(Bm + (size_t)(k0 + br) * N + bn + bc, &Bs[br * 68 + bc]);
    }
    wait_async0();
    __syncthreads();

#pragma unroll
    for (int kk = 0; kk < 32; kk += 4) {
      const int kA = kk + kh;
      v2f a0, a1;
      a0.x = As[(32 * wave + rowl) * 36 + kA];
      a0.y = As[(32 * wave + rowl) * 36 + kA + 1];
      a1.x = As[(32 * wave + 16 + rowl) * 36 + kA];
      a1.y = As[(32 * wave + 16 + rowl) * 36 + kA + 1];
#pragma unroll
      for (int ct = 0; ct < 4; ct++) {
        v2f b;
        b.x = Bs[kA * 68 + ct * 16 + rowl];
        b.y = Bs[(kA + 1) * 68 + ct * 16 + rowl];
        acc0[ct] = wmma_f32_k4(a0, b, acc0[ct]);
        acc1[ct] = wmma_f32_k4(a1, b, acc1[ct]);
      }
    }
    __syncthreads();

    if (k0 + 32 < K) {  // prefetch next K-panel -> global_prefetch_b8
      __builtin_prefetch(A + (size_t)(bm + tid) * K + k0 + 32, 0, 3);
      __builtin_prefetch(Bm + (size_t)(k0 + 32 + (tid >> 2)) * N + bn, 0, 3);
    }
  }

  // C/D layout: element (M = r + 8*(lane>>4), N = lane&15) in acc[ct][r]
  const int mrow0 = bm + 32 * wave + ((lane >> 4) << 3);
#pragma unroll
  for (int ct = 0; ct < 4; ct++)
#pragma unroll
    for (int r = 0; r < 8; r++) {
      Cm[(size_t)(mrow0 + r) * N + bn + ct * 16 + rowl] = acc0[ct][r];
      Cm[(size_t)(mrow0 + 16 + r) * N + bn + ct * 16 + rowl] = acc1[ct][r];
    }
}

// ---------------------------------------------------------------------------
// RoPE on q in place. q layout: (B,T,NH,HD). One thread per (b,t,h,j<64) pair.
// ---------------------------------------------------------------------------
__global__ __launch_bounds__(256) void rope_q_kernel(float* __restrict__ qw) {
  const int idx = blockIdx.x * 256 + threadIdx.x;  // B*T*NH*64 = 4194304
  const int j = idx & 63;
  const int h = (idx >> 6) & (NH - 1);
  const int t = (idx >> 11) & (T_ - 1);
  const int b = idx >> 21;
  float* p = qw + (size_t)(b * T_ + t) * C_ + h * HD;
  // inv_freq = 10000^(-j/64) = exp(-j*ln(10000)/64)
  const float inv = __expf(-0.14391157f * (float)j);
  const float ang = (float)(P_ + t) * inv;
  float sn, cs;
  sincosf(ang, &sn, &cs);
  const float x1 = p[j], x2 = p[j + 64];
  p[j] = x1 * cs - x2 * sn;
  p[j + 64] = x1 * sn + x2 * cs;
}

// ---------------------------------------------------------------------------
// Build fp32 K/V caches (B,NKV,S,HD):
//   j <  P_ : K = dequant(pk_i8 * pk_s), V = pv
//   j >= P_ : K = RoPE(k from kv projection), V = v from kv projection
// ---------------------------------------------------------------------------
__global__ __launch_bounds__(256) void build_cache_kernel(
    const float* __restrict__ kvw, const int8_t* __restrict__ pk_i8,
    const float* __restrict__ pk_s, const float* __restrict__ pv,
    float* __restrict__ kcat, float* __restrict__ vout) {
  const int idx = blockIdx.x * 256 + threadIdx.x;  // B*NKV*S*HD = 4194304
  const int d = idx & (HD - 1);
  const int j = (idx >> 7) & (S_ - 1);
  const int kvh = (idx >> 18) & (NKV - 1);
  const int b = idx >> 21;
  if (j < P_) {
    const size_t pidx = (size_t)(b * NKV + kvh) * P_ + j;
    kcat[idx] = (float)pk_i8[pidx * HD + d] * pk_s[pidx];
    vout[idx] = pv[pidx * HD + d];
  } else {
    const int t = j - P_;
    const float* kvrow = kvw + (size_t)(b * T_ + t) * KVC;
    vout[idx] = kvrow[NKV * HD + kvh * HD + d];  // v slot
    const int j2 = d & 63;
    const float inv = __expf(-0.14391157f * (float)j2);
    const float ang = (float)(P_ + t) * inv;
    float sn, cs;
    sincosf(ang, &sn, &cs);
    const float* kp = kvrow + kvh * HD;  // k slot
    const float x1 = kp[j2], x2 = kp[j2 + 64];
    kcat[idx] = (d < 64) ? (x1 * cs - x2 * sn) : (x1 * sn + x2 * cs);
  }
}

// ---------------------------------------------------------------------------
// Requantize concatenated K: one wave per row of 128, shfl amax reduction.
// k_i8 emitted as float values (harness flattens tuple to fp32).
// ---------------------------------------------------------------------------
__global__ __launch_bounds__(256) void quantize_rows_kernel(
    const float* __restrict__ kcat, float* __restrict__ ki8_out,
    float* __restrict__ ks_out) {
  const int row = blockIdx.x * 8 + (threadIdx.x >> 5);  // B*NKV*S = 32768 rows
  const int lane = threadIdx.x & 31;
  const float4 v =
      *reinterpret_cast<const float4*>(kcat + (size_t)row * HD + lane * 4);
  float am = fmaxf(fmaxf(fabsf(v.x), fabsf(v.y)), fmaxf(fabsf(v.z), fabsf(v.w)));
  am = fmaxf(am, __shfl_xor(am, 1, 32));
  am = fmaxf(am, __shfl_xor(am, 2, 32));
  am = fmaxf(am, __shfl_xor(am, 4, 32));
  am = fmaxf(am, __shfl_xor(am, 8, 32));
  am = fmaxf(am, __shfl_xor(am, 16, 32));
  const float scale = fmaxf(am * (1.0f / 127.0f), 1e-8f);
  const float rs = 1.0f / scale;
  float4 o;
  o.x = fminf(fmaxf(rintf(v.x * rs), -127.f), 127.f);
  o.y = fminf(fmaxf(rintf(v.y * rs), -127.f), 127.f);
  o.z = fminf(fmaxf(rintf(v.z * rs), -127.f), 127.f);
  o.w = fminf(fmaxf(rintf(v.w * rs), -127.f), 127.f);
  *reinterpret_cast<float4*>(ki8_out + (size_t)row * HD + lane * 4) = o;
  if (lane == 0) ks_out[row] = scale;
}

// ---------------------------------------------------------------------------
// Flash attention, fp32 WMMA. Block = 128 (4 waves); wave owns one 16-row
// query tile. Q fragments (K=128 -> 32 v2f) held in registers; score tile
// round-trips padded LDS to convert D-layout -> A-layout for P*V.
// Causal: query qi sees keys j <= P_ + qi; only the diagonal tile is masked.
// ---------------------------------------------------------------------------
__global__ __launch_bounds__(128) void attn_wmma_kernel(
    const float* __restrict__ q, const float* __restrict__ kcat,
    const float* __restrict__ vcat, float* __restrict__ aout) {
  __shared__ float Pt[4][16 * 17];
  const int b = blockIdx.z, h = blockIdx.y;
  const int wave = threadIdx.x >> 5, lane = threadIdx.x & 31;
  const int qtile = blockIdx.x * 4 + wave;  // 0..63
  const int kvh = h >> 2;                   // rep = NH/NKV = 4
  const int rowl = lane & 15;
  const int kh = (lane >> 4) << 1;
  const float* qb = q + (size_t)(b * T_ + qtile * 16) * C_ + h * HD;
  const float* kb = kcat + (size_t)(b * NKV + kvh) * S_ * HD;
  const float* vb = vcat + (size_t)(b * NKV + kvh) * S_ * HD;
  float* Pw = &Pt[wave][0];

  const float sm_scale = 0.08838834764831845f;  // 1/sqrt(128)
  v2f qa[32];
#pragma unroll
  for (int kt = 0; kt < 32; kt++) {
    const float2 t =
        *reinterpret_cast<const float2*>(qb + (size_t)rowl * C_ + kt * 4 + kh);
    qa[kt].x = t.x * sm_scale;
    qa[kt].y = t.y * sm_scale;
  }

  v8f acc[8] = {};
  float mrow[8], lrow[8];
#pragma unroll
  for (int r = 0; r < 8; r++) { mrow[r] = -3.0e38f; lrow[r] = 0.f; }

  const int last_nt = 64 + qtile;  // inclusive; later tiles fully masked
  for (int nt = 0; nt <= last_nt; nt++) {
    if (nt < last_nt) {  // prefetch next K/V tiles -> global_prefetch_b8
      const size_t nrow = (size_t)((nt + 1) * 16 + rowl) * HD;
      __builtin_prefetch(kb + nrow + ((lane >> 4) << 6), 0, 3);
      __builtin_prefetch(vb + nrow + ((lane >> 4) << 6), 0, 3);
    }
    // ---- scores S = (q * 1/sqrt(hd)) @ K^T, 16x16 tile, K-dim 128 ----
    v8f s = {};
    const float* kt0 = kb + (size_t)(nt * 16 + rowl) * HD;
#pragma unroll
    for (int kt = 0; kt < 32; kt++) {
      const float2 t = *reinterpret_cast<const float2*>(kt0 + kt * 4 + kh);
      v2f bf;
      bf.x = t.x;
      bf.y = t.y;
      s = wmma_f32_k4(qa[kt], bf, s);
    }
    if (nt == last_nt) {  // diagonal tile: key nn allowed iff nn <= M
#pragma unroll
      for (int r = 0; r < 8; r++) {
        const int M = r + ((lane >> 4) << 3);
        s[r] = (rowl <= M) ? s[r] : -1.0e30f;
      }
    }
    // ---- online softmax (rows live in 16-lane halves) ----
    float pr[8];
#pragma unroll
    for (int r = 0; r < 8; r++) {
      float x = s[r];
      x = fmaxf(x, __shfl_xor(x, 1, 32));
      x = fmaxf(x, __shfl_xor(x, 2, 32));
      x = fmaxf(x, __shfl_xor(x, 4, 32));
      x = fmaxf(x, __shfl_xor(x, 8, 32));
      const float nm = fmaxf(mrow[r], x);
      const float sc = __expf(mrow[r] - nm);
      mrow[r] = nm;
      float p = __expf(s[r] - nm);
      pr[r] = p;
      float ps = p;
      ps += __shfl_xor(ps, 1, 32);
      ps += __shfl_xor(ps, 2, 32);
      ps += __shfl_xor(ps, 4, 32);
      ps += __shfl_xor(ps, 8, 32);
      lrow[r] = lrow[r] * sc + ps;
#pragma unroll
      for (int dt = 0; dt < 8; dt++) acc[dt][r] *= sc;
    }
    // ---- D-layout -> A-layout through LDS (wave-private, hw in-order) ----
#pragma unroll
    for (int r = 0; r < 8; r++)
      Pw[(r + ((lane >> 4) << 3)) * 17 + rowl] = pr[r];
    // ---- acc += P @ V (16 keys x 128 dims) ----
#pragma unroll
    for (int kk = 0; kk < 4; kk++) {
      const int kA = kk * 4 + kh;
      v2f a;
      a.x = Pw[rowl * 17 + kA];
      a.y = Pw[rowl * 17 + kA + 1];
      const float* vr = vb + (size_t)(nt * 16 + kA) * HD;
#pragma unroll
      for (int dt = 0; dt < 8; dt++) {
        v2f bf;
        bf.x = vr[dt * 16 + rowl];
        bf.y = vr[HD + dt * 16 + rowl];
        acc[dt] = wmma_f32_k4(a, bf, acc[dt]);
      }
    }
  }

  // ---- normalize and store in (b, t, h, d) order for the Wo GEMM ----
  float inv[8];
#pragma unroll
  for (int r = 0; r < 8; r++) inv[r] = 1.0f / lrow[r];
  float* ob = aout + (size_t)(b * T_ + qtile * 16) * C_ + h * HD;
#pragma unroll
  for (int dt = 0; dt < 8; dt++)
#pragma unroll
    for (int r = 0; r < 8; r++)
      ob[(size_t)(r + ((lane >> 4) << 3)) * C_ + dt * 16 + rowl] =
          acc[dt][r] * inv[r];
}

// ---------------------------------------------------------------------------
extern "C" void kernel_launch(void* const* d_in, const int* in_sizes, int n_in,
                              void* d_out, int out_size, void* d_ws,
                              size_t ws_size, hipStream_t stream) {
  const float* x = (const float*)d_in[0];
  const float* Wq = (const float*)d_in[1];
  const float* Wkv = (const float*)d_in[2];
  const float* Wo = (const float*)d_in[3];
  const int8_t* pk_i8 = (const int8_t*)d_in[4];
  const float* pk_s = (const float*)d_in[5];
  const float* pv = (const float*)d_in[6];
  // d_in[7] = start_pos (== P_, compile-time constant here)

  // d_out: tuple flattened as fp32 in return order (out, k_i8, k_s, v)
  float* out = (float*)d_out;
  float* out_ki8 = out + (size_t)B_ * T_ * C_;            // 8388608
  float* out_ks = out_ki8 + (size_t)B_ * NKV * S_ * HD;   // +4194304
  float* out_v = out_ks + (size_t)B_ * NKV * S_;          // +32768

  // workspace: q(32MB) + kv(16MB) + kcat(16MB) + attn(32MB) = 96MB
  float* ws = (float*)d_ws;
  float* q_ws = ws;
  float* kv_ws = q_ws + (size_t)B_ * T_ * C_;
  float* kcat_ws = kv_ws + (size_t)B_ * T_ * KVC;
  float* attn_ws = kcat_ws + (size_t)B_ * NKV * S_ * HD;

  const int Mrows = B_ * T_;  // 2048

  // 1) q = x @ Wq
  gemm_f32_wmma<<<dim3(C_ / 64, Mrows / 128), 128, 0, stream>>>(x, Wq, q_ws,
                                                                Mrows, C_, C_);
  // 2) kv = x @ Wkv
  gemm_f32_wmma<<<dim3(KVC / 64, Mrows / 128), 128, 0, stream>>>(
      x, Wkv, kv_ws, Mrows, KVC, C_);
  // 3) RoPE(q) in place
  rope_q_kernel<<<(B_ * T_ * NH * 64) / 256, 256, 0, stream>>>(q_ws);
  // 4) K/V caches: dequant old + RoPE new k; v straight through
  build_cache_kernel<<<(B_ * NKV * S_ * HD) / 256, 256, 0, stream>>>(
      kv_ws, pk_i8, pk_s, pv, kcat_ws, out_v);
  // 5) requantize concatenated K -> k_i8 (as float), k_s
  quantize_rows_kernel<<<(B_ * NKV * S_) / 8, 256, 0, stream>>>(
      kcat_ws, out_ki8, out_ks);
  // 6) flash attention (reads v from d_out region written in step 4)
  attn_wmma_kernel<<<dim3(16, NH, B_), 128, 0, stream>>>(q_ws, kcat_ws, out_v,
                                                         attn_ws);
  // 7) out = attn @ Wo
  gemm_f32_wmma<<<dim3(C_ / 64, Mrows / 128), 128, 0, stream>>>(
      attn_ws, Wo, out, Mrows, C_, C_);
}